// DL_BBBP_onlygraph_88974542504172
// MI455X (gfx1250) — compile-verified
//
#include <hip/hip_runtime.h>
#include <math.h>

typedef float v2f __attribute__((ext_vector_type(2)));
typedef float v8f __attribute__((ext_vector_type(8)));

#define CDIV(a, b) (((a) + (b) - 1) / (b))

// ---------------------------------------------------------------------------
// Degree / normalization
// ---------------------------------------------------------------------------
__global__ void degree_init_kernel(float* deg, int n) {
    int i = blockIdx.x * blockDim.x + threadIdx.x;
    if (i < n) deg[i] = 1.0f;  // self-loop
}

__global__ void degree_accum_kernel(const int* __restrict__ dst, float* deg, int e) {
    int i = blockIdx.x * blockDim.x + threadIdx.x;
    if (i < e) atomicAdd(&deg[dst[i]], 1.0f);
}

__global__ void dinv_kernel(float* deg, int n) {
    int i = blockIdx.x * blockDim.x + threadIdx.x;
    if (i < n) deg[i] = rsqrtf(deg[i]);  // deg >= 1 always (self-loop)
}

// ---------------------------------------------------------------------------
// fp32 WMMA GEMM:  C[M,N] = A[M,K] @ W[K,N] (+bias)(+relu)
// One wave computes a 16x16 tile with V_WMMA_F32_16X16X4_F32.
//
// Branchless hot loop: OOB A-rows / B-cols are CLAMPED (their products land
// only in accumulator rows/cols that are never stored), so no masks/selects
// are needed inside the K loop. Only the one-shot K%4 tail uses selects.
// ---------------------------------------------------------------------------
__global__ void gemm_wmma_f32(const float* __restrict__ A, const float* __restrict__ W,
                              const float* __restrict__ bias, float* __restrict__ C,
                              int M, int N, int K, int relu) {
    const int wave = blockIdx.x * (blockDim.x >> 5) + (threadIdx.x >> 5);
    const int lane = threadIdx.x & 31;
    const int ntiles = (N + 15) >> 4;
    const int mtiles = (M + 15) >> 4;
    if (wave >= mtiles * ntiles) return;          // wave-uniform: EXEC stays full
    const int m0 = (wave / ntiles) << 4;
    const int n0 = (wave % ntiles) << 4;

    const int lrow = lane & 15;
    const int lhi  = lane >> 4;                   // K-pair select within 4-wide step

    const int arow = min(m0 + lrow, M - 1);       // clamped; OOB rows never stored
    const int bcol = min(n0 + lrow, N - 1);       // clamped; OOB cols never stored

    const float* Ap = A + (size_t)arow * K + 2 * lhi;
    const float* Bp = W + (size_t)(2 * lhi) * N + bcol;
    const size_t bstep = (size_t)4 * N;

    v8f acc = {0.f, 0.f, 0.f, 0.f, 0.f, 0.f, 0.f, 0.f};

    const int kmain = K & ~3;
#pragma unroll 4
    for (int k = 0; k < kmain; k += 4) {
        v2f af, bf;
        af.x = Ap[0];
        af.y = Ap[1];
        bf.x = Bp[0];
        bf.y = Bp[N];
        Ap += 4;
        Bp += bstep;
        acc = __builtin_amdgcn_wmma_f32_16x16x4_f32(false, af, false, bf,
                                                    (short)0, acc, false, false);
    }
    if (K & 3) {                                  // one-shot tail, branchless
        const int krem = K - kmain;               // 1..3
        const int ka = 2 * lhi;
        const bool v0 = ka < krem;
        const bool v1 = (ka + 1) < krem;
        const float* a0p = v0 ? Ap     : A;       // pre-selected safe addresses
        const float* a1p = v1 ? Ap + 1 : A;
        const float* b0p = v0 ? Bp     : W;
        const float* b1p = v1 ? Bp + N : W;
        const float a0 = *a0p, a1 = *a1p, b0 = *b0p, b1 = *b1p;
        v2f af, bf;
        af.x = v0 ? a0 : 0.f;
        af.y = v1 ? a1 : 0.f;
        bf.x = v0 ? b0 : 0.f;
        bf.y = v1 ? b1 : 0.f;
        acc = __builtin_amdgcn_wmma_f32_16x16x4_f32(false, af, false, bf,
                                                    (short)0, acc, false, false);
    }

    if (n0 + lrow >= N) return;                   // whole column OOB: nothing to store
    const int scol = n0 + lrow;
    const float bv = bias ? bias[scol] : 0.f;
#pragma unroll
    for (int i = 0; i < 8; ++i) {
        const int row = m0 + (lhi << 3) + i;      // C/D layout: lane-half = M-half
        if (row < M) {
            float v = acc[i] + bv;
            if (relu) v = fmaxf(v, 0.f);
            C[(size_t)row * N + scol] = v;
        }
    }
}

// ---------------------------------------------------------------------------
// GCN aggregation
// ---------------------------------------------------------------------------
__global__ void gcn_self_init_kernel(const float* __restrict__ h, const float* __restrict__ dinv,
                                     float* __restrict__ out, int n, int F) {
    size_t idx = (size_t)blockIdx.x * blockDim.x + threadIdx.x;
    size_t total = (size_t)n * F;
    if (idx < total) {
        int node = (int)(idx / F);
        float di = dinv[node];
        out[idx] = h[idx] * di * di;              // self-loop contribution
    }
}

__global__ void gcn_scatter_kernel(const float* __restrict__ h, const float* __restrict__ dinv,
                                   const int* __restrict__ src, const int* __restrict__ dst,
                                   float* __restrict__ out, int E, int F) {
    const int wave = blockIdx.x * (blockDim.x >> 5) + (threadIdx.x >> 5);
    const int lane = threadIdx.x & 31;
    if (wave >= E) return;
    const int s = src[wave];
    const int d = dst[wave];
    const float w = dinv[s] * dinv[d];
    const float* hs = h + (size_t)s * F;
    float* od = out + (size_t)d * F;
    for (int f = lane; f < F; f += 32)
        atomicAdd(&od[f], hs[f] * w);
}

__global__ void bias_relu_kernel(float* __restrict__ out, const float* __restrict__ bias,
                                 int n, int F) {
    size_t idx = (size_t)blockIdx.x * blockDim.x + threadIdx.x;
    size_t total = (size_t)n * F;
    if (idx < total) {
        int f = (int)(idx % F);
        out[idx] = fmaxf(out[idx] + bias[f], 0.f);
    }
}

// ---------------------------------------------------------------------------
// Global max pool per graph (int-pattern atomicMax; inputs are >= 0 post-ReLU)
// ---------------------------------------------------------------------------
__global__ void pool_init_kernel(int* g, size_t total) {
    size_t idx = (size_t)blockIdx.x * blockDim.x + threadIdx.x;
    if (idx < total) g[idx] = 0xFF7FFFFF;         // bit pattern of -FLT_MAX
}

__global__ void pool_max_kernel(const float* __restrict__ x, const int* __restrict__ batch,
                                int* __restrict__ g, int n, int F) {
    const int wave = blockIdx.x * (blockDim.x >> 5) + (threadIdx.x >> 5);
    const int lane = threadIdx.x & 31;
    if (wave >= n) return;
    const int gi = batch[wave];
    const float* xr = x + (size_t)wave * F;
    int* gr = g + (size_t)gi * F;
    for (int f = lane; f < F; f += 32)
        atomicMax(&gr[f], __float_as_int(xr[f]));
}

// ---------------------------------------------------------------------------
// Final 512 -> 1 dot product + sigmoid, one wave per row
// ---------------------------------------------------------------------------
__global__ void final_sigmoid_kernel(const float* __restrict__ c, const float* __restrict__ Wo,
                                     const float* __restrict__ bo, float* __restrict__ out,
                                     int M, int K) {
    const int wave = blockIdx.x * (blockDim.x >> 5) + (threadIdx.x >> 5);
    const int lane = threadIdx.x & 31;
    if (wave >= M) return;
    const float* cr = c + (size_t)wave * K;
    float s = 0.f;
    for (int k = lane; k < K; k += 32) s += cr[k] * Wo[k];
#pragma unroll
    for (int off = 16; off >= 1; off >>= 1) s += __shfl_xor(s, off, 32);
    if (lane == 0) out[wave] = 1.f / (1.f + expf(-(s + bo[0])));
}

// ---------------------------------------------------------------------------
// Host launcher
// ---------------------------------------------------------------------------
extern "C" void kernel_launch(void* const* d_in, const int* in_sizes, int n_in,
                              void* d_out, int out_size, void* d_ws, size_t ws_size,
                              hipStream_t stream) {
    const float* x     = (const float*)d_in[0];
    const int*   edge  = (const int*)d_in[1];
    const int*   batch = (const int*)d_in[2];
    const float* W1 = (const float*)d_in[3];  const float* b1 = (const float*)d_in[4];
    const float* W2 = (const float*)d_in[5];  const float* b2 = (const float*)d_in[6];
    const float* W3 = (const float*)d_in[7];  const float* b3 = (const float*)d_in[8];
    const float* Wg1 = (const float*)d_in[9];  const float* bg1 = (const float*)d_in[10];
    const float* Wg2 = (const float*)d_in[11]; const float* bg2 = (const float*)d_in[12];
    const float* Wf1 = (const float*)d_in[13]; const float* bf1 = (const float*)d_in[14];
    const float* Wf2 = (const float*)d_in[15]; const float* bf2 = (const float*)d_in[16];
    const float* Wo  = (const float*)d_in[17]; const float* bo  = (const float*)d_in[18];

    // Derive dims from in_sizes / out_size
    const int F1 = in_sizes[4];                 // 78
    const int F2 = in_sizes[6];                 // 156
    const int F3 = in_sizes[8];                 // 312
    const int H1 = in_sizes[10];                // 1024
    const int H2 = in_sizes[12];                // 128
    const int H3 = in_sizes[14];                // 1024
    const int H4 = in_sizes[16];                // 512
    const int XD = in_sizes[3] / F1;            // 78
    const int Nn = in_sizes[0] / XD;            // 50000
    const int E  = in_sizes[1] / 2;             // 800000
    const int Gn = out_size;                    // 2048
    const int* src = edge;
    const int* dst = edge + E;

    // Workspace carve-out (256B aligned)
    char* ws = (char*)d_ws;
    size_t off = 0;
    auto alloc = [&](size_t bytes) -> void* {
        void* p = ws + off;
        off = (off + bytes + 255) & ~(size_t)255;
        return p;
    };
    float* dinv = (float*)alloc((size_t)Nn * sizeof(float));
    float* bufA = (float*)alloc((size_t)Nn * F3 * sizeof(float));   // 62.4 MB
    float* bufB = (float*)alloc((size_t)Nn * F3 * sizeof(float));   // 62.4 MB

    // Dense-phase buffers alias the GCN buffers (free by then, non-overlapping)
    float* Gbuf = bufA;                                     // [Gn,F3]  (2.6 MB)
    float* D1   = bufB;                                     // [Gn,H1]  (8 MB)
    float* D2   = bufA + (size_t)(4  * 1024 * 1024) / 4;    // [Gn,H2]  (1 MB)
    float* D3   = bufB + (size_t)(16 * 1024 * 1024) / 4;    // [Gn,H3]  (8 MB)
    float* D4   = bufA + (size_t)(16 * 1024 * 1024) / 4;    // [Gn,H4]  (4 MB)

    const int T = 256;  // 8 waves per block

    // --- degree -> dinv ---
    degree_init_kernel<<<CDIV(Nn, T), T, 0, stream>>>(dinv, Nn);
    degree_accum_kernel<<<CDIV(E, T), T, 0, stream>>>(dst, dinv, E);
    dinv_kernel<<<CDIV(Nn, T), T, 0, stream>>>(dinv, Nn);

    // --- one GCN layer: hbuf = xin@W; obuf = relu(scatter(norm * gather(hbuf)) + b) ---
    auto gcn_layer = [&](const float* xin, const float* W, const float* b,
                         int Fin, int Fout, float* hbuf, float* obuf) {
        int waves = CDIV(Nn, 16) * CDIV(Fout, 16);
        gemm_wmma_f32<<<CDIV(waves, 8), T, 0, stream>>>(xin, W, nullptr, hbuf, Nn, Fout, Fin, 0);
        size_t tot = (size_t)Nn * Fout;
        gcn_self_init_kernel<<<(int)CDIV(tot, T), T, 0, stream>>>(hbuf, dinv, obuf, Nn, Fout);
        gcn_scatter_kernel<<<CDIV(E, 8), T, 0, stream>>>(hbuf, dinv, src, dst, obuf, E, Fout);
        bias_relu_kernel<<<(int)CDIV(tot, T), T, 0, stream>>>(obuf, b, Nn, Fout);
    };

    gcn_layer(x,    W1, b1, XD, F1, bufA, bufB);   // layer 1: x -> bufB
    gcn_layer(bufB, W2, b2, F1, F2, bufA, bufB);   // layer 2
    gcn_layer(bufB, W3, b3, F2, F3, bufA, bufB);   // layer 3: features in bufB

    // --- global max pool per graph ---
    {
        size_t tot = (size_t)Gn * F3;
        pool_init_kernel<<<(int)CDIV(tot, T), T, 0, stream>>>((int*)Gbuf, tot);
        pool_max_kernel<<<CDIV(Nn, 8), T, 0, stream>>>(bufB, batch, (int*)Gbuf, Nn, F3);
    }

    // --- dense head (WMMA GEMMs with fused bias / relu) ---
    auto dense = [&](const float* in, const float* W, const float* b, float* o,
                     int M, int Nc, int K, int relu) {
        int waves = CDIV(M, 16) * CDIV(Nc, 16);
        gemm_wmma_f32<<<CDIV(waves, 8), T, 0, stream>>>(in, W, b, o, M, Nc, K, relu);
    };
    dense(Gbuf, Wg1, bg1, D1, Gn, H1, F3, 1);   // 312 -> 1024, relu
    dense(D1,   Wg2, bg2, D2, Gn, H2, H1, 0);   // 1024 -> 128
    dense(D2,   Wf1, bf1, D3, Gn, H3, H2, 1);   // 128 -> 1024, relu
    dense(D3,   Wf2, bf2, D4, Gn, H4, H3, 1);   // 1024 -> 512, relu

    // --- final 512 -> 1 + sigmoid ---
    final_sigmoid_kernel<<<CDIV(Gn, 8), T, 0, stream>>>(D4, Wo, bo, (float*)d_out, Gn, H4);
}